// CPDecodeForExport_66864050864553
// MI455X (gfx1250) — compile-verified
//
#include <hip/hip_runtime.h>
#include <hip/hip_bf16.h>
#include <math.h>

// ---------------- model constants (from reference) ----------------
constexpr int NL  = 5;
constexpr int HID = 1024;
constexpr int H   = 16;
constexpr int KV  = 8;
constexpr int D   = 128;
constexpr int FF  = 3072;
constexpr int LCTX = 8192;
constexpr float EPS = 1e-6f;

typedef float v2f __attribute__((ext_vector_type(2)));
typedef float v8f __attribute__((ext_vector_type(8)));

// ---------------- helpers ----------------
__device__ inline float wave_reduce_sum(float v) {
  #pragma unroll
  for (int off = 16; off > 0; off >>= 1) v += __shfl_xor(v, off, 32);
  return v;
}
__device__ inline float wave_reduce_max(float v) {
  #pragma unroll
  for (int off = 16; off > 0; off >>= 1) v = fmaxf(v, __shfl_xor(v, off, 32));
  return v;
}

// ---------------- tiny copy ----------------
__global__ void copy_kernel(const float* __restrict__ src, float* __restrict__ dst, int n) {
  int i = blockIdx.x * 256 + threadIdx.x;
  if (i < n) dst[i] = src[i];
}

// ---------------- RMS norm (1 block, 256 threads) ----------------
__global__ void rmsnorm_kernel(const float* __restrict__ x, const float* __restrict__ w,
                               float* __restrict__ out, int n) {
  __shared__ float red[8];
  __shared__ float rs_sh;
  int t = threadIdx.x;
  float ss = 0.f;
  for (int j = t; j < n; j += 256) { float v = x[j]; ss += v * v; }
  ss = wave_reduce_sum(ss);
  if ((t & 31) == 0) red[t >> 5] = ss;
  __syncthreads();
  if (t == 0) {
    float s = 0.f;
    #pragma unroll
    for (int k = 0; k < 8; ++k) s += red[k];
    rs_sh = rsqrtf(s / (float)n + EPS);
  }
  __syncthreads();
  float rs = rs_sh;
  for (int j = t; j < n; j += 256) out[j] = x[j] * rs * w[j];
}

// ---------------- GEMV: y[r] = dot(W[r,:], x), wave per row, float4 streams ----------------
__global__ void gemv_kernel(const float* __restrict__ W, const float* __restrict__ x,
                            float* __restrict__ y, int R, int K) {
  int wave = (int)((blockIdx.x * blockDim.x + threadIdx.x) >> 5);
  int lane = threadIdx.x & 31;
  if (wave >= R) return;
  const float* wr = W + (size_t)wave * K;
  float acc = 0.f;
  for (int k = lane * 4; k < K; k += 128) {
    float4 a = *(const float4*)(wr + k);
    float4 b = *(const float4*)(x + k);
    acc += a.x * b.x + a.y * b.y + a.z * b.z + a.w * b.w;
  }
  acc = wave_reduce_sum(acc);
  if (lane == 0) y[wave] = acc;
}

__global__ void gemv_residual_kernel(const float* __restrict__ W, const float* __restrict__ x,
                                     float* __restrict__ y, int R, int K) {
  int wave = (int)((blockIdx.x * blockDim.x + threadIdx.x) >> 5);
  int lane = threadIdx.x & 31;
  if (wave >= R) return;
  const float* wr = W + (size_t)wave * K;
  float acc = 0.f;
  for (int k = lane * 4; k < K; k += 128) {
    float4 a = *(const float4*)(wr + k);
    float4 b = *(const float4*)(x + k);
    acc += a.x * b.x + a.y * b.y + a.z * b.z + a.w * b.w;
  }
  acc = wave_reduce_sum(acc);
  if (lane == 0) y[wave] += acc;
}

// fused gate/up GEMV: act[r] = silu(gate_r . x) * (up_r . x)
__global__ void gemv_gateup_kernel(const float* __restrict__ G, const float* __restrict__ U,
                                   const float* __restrict__ x, float* __restrict__ act,
                                   int R, int K) {
  int wave = (int)((blockIdx.x * blockDim.x + threadIdx.x) >> 5);
  int lane = threadIdx.x & 31;
  if (wave >= R) return;
  const float* gr = G + (size_t)wave * K;
  const float* ur = U + (size_t)wave * K;
  float ag = 0.f, au = 0.f;
  for (int k = lane * 4; k < K; k += 128) {
    float4 b = *(const float4*)(x + k);
    float4 a = *(const float4*)(gr + k);
    ag += a.x * b.x + a.y * b.y + a.z * b.z + a.w * b.w;
    float4 c = *(const float4*)(ur + k);
    au += c.x * b.x + c.y * b.y + c.z * b.z + c.w * b.w;
  }
  ag = wave_reduce_sum(ag);
  au = wave_reduce_sum(au);
  if (lane == 0) {
    float s = ag / (1.f + expf(-ag));   // silu
    act[wave] = s * au;
  }
}

// ---------------- per-head q/k RMS norm + RoPE (grid = 24 heads, 128 threads) ----------------
__global__ void qknorm_rope_kernel(const float* __restrict__ qraw, const float* __restrict__ kraw,
                                   const float* __restrict__ qn, const float* __restrict__ kn,
                                   const int* __restrict__ cache_pos,
                                   float* __restrict__ qout, float* __restrict__ kout) {
  __shared__ float xn[128];
  __shared__ float red[4];
  __shared__ float rs_sh;
  int b = blockIdx.x, t = threadIdx.x;
  const float* x; const float* w; float* out;
  if (b < 16) { x = qraw + b * 128; w = qn; out = qout + b * 128; }
  else        { x = kraw + (b - 16) * 128; w = kn; out = kout + (b - 16) * 128; }
  float v = x[t];
  float ss = wave_reduce_sum(v * v);
  if ((t & 31) == 0) red[t >> 5] = ss;
  __syncthreads();
  if (t == 0) rs_sh = rsqrtf((red[0] + red[1] + red[2] + red[3]) / 128.f + EPS);
  __syncthreads();
  xn[t] = v * rs_sh * w[t];
  __syncthreads();
  float pos = (float)cache_pos[0];
  int d2 = (t < 64) ? t : (t - 64);
  float freq = powf(1000000.0f, -(float)d2 / 64.0f);   // THETA^(-(2i)/D)
  float ang = pos * freq;
  float c = cosf(ang), s = sinf(ang);
  float o = (t < 64) ? (xn[t] * c - xn[t + 64] * s)
                     : (xn[t] * c + xn[t - 64] * s);
  out[t] = o;
}

// ---------------- fused attention scores + K-cache copy-out (WMMA f32 16x16x4) ----------------
// 4 waves/block; each wave owns one (kv-head g, 16-position) tile; all 4 tiles of a
// block share the same kv head g.  Waves stream their 16x128 K tile from past_keys
// (substituting the fresh RoPE'd k at `pos`), write it to out_pk (fused copy-out),
// and stage it in LDS.  A zero-padded Q tile (rows 0-1 = the two GQA query heads,
// rows 2-15 = 0) is built once per block in LDS so that both WMMA fragments come
// from unconditional ds_load_b64 with uniform control flow (EXEC all-ones).
__global__ void score_copyk_kernel(const float* __restrict__ pastK,   // KV*L*D (layer base)
                                   const float* __restrict__ newk,    // KV*D
                                   const float* __restrict__ q,       // H*D
                                   const float* __restrict__ mask,    // L
                                   const int* __restrict__ cache_pos,
                                   float* __restrict__ outK,          // KV*L*D (layer base)
                                   float* __restrict__ scores) {      // H*L
  __shared__ float kt[4][16 * 128];                  // 32 KB
  __shared__ float qpad[16 * 128];                   // 8 KB, shared by the block
  int t = threadIdx.x;                               // 128 threads
  int wv = t >> 5, lane = t & 31;
  int tile = blockIdx.x * 4 + wv;                    // 0..4095
  int g  = tile >> 9;                                // kv head (uniform per block)
  int j0 = (tile & 511) << 4;                        // position tile start
  int pos = cache_pos[0];
  const float* Kg = pastK + ((size_t)g * LCTX + j0) * D;
  float*       Og = outK  + ((size_t)g * LCTX + j0) * D;
  float* kl = kt[wv];

  // Zero-padded A-matrix tile: rows 0,1 = q heads g*2, g*2+1; rows 2..15 = 0.
  for (int idx = t; idx < 16 * 128; idx += 128) {
    int row = idx >> 7, col = idx & 127;
    qpad[idx] = (row < 2) ? q[(g * 2 + row) * D + col] : 0.f;
  }

  // 16 rows x 128 cols = 512 float4 per tile; fully coalesced per wave.
  for (int idx = lane; idx < 512; idx += 32) {
    int row = idx >> 5;
    int c4  = (idx & 31) << 2;
    const float* src = ((j0 + row) == pos) ? (newk + g * D + c4) : (Kg + (size_t)row * D + c4);
    float4 v = *(const float4*)src;
    *(float4*)(Og + (size_t)row * D + c4) = v;       // fused cache copy-out
    *(float4*)(kl + row * 128 + c4) = v;
  }
  __syncthreads();

  // WMMA over K=128 in steps of 4: ds_load_b64 (A) + ds_load_b64 (B) + v_wmma.
  v8f acc = {};
  int li   = lane & 15;
  int half = lane >> 4;
  const float* arow = qpad + li * 128 + half * 2;    // A: lanes 0-15 K=kk,kk+1; 16-31 K=kk+2,kk+3
  const float* brow = kl   + li * 128 + half * 2;    // B[k][n] = Ktile[n][k] (mirrored layout)
  #pragma unroll 4
  for (int kk = 0; kk < 128; kk += 4) {
    v2f a = *(const v2f*)(arow + kk);
    v2f b = *(const v2f*)(brow + kk);
    acc = __builtin_amdgcn_wmma_f32_16x16x4_f32(false, a, false, b, (short)0, acc, false, false);
  }
  // D fragment: VGPR r, lanes 0-15 = [M=r][N=lane]; rows 0,1 are the real heads.
  const float sc = 0.08838834764831845f;             // D^-0.5
  if (lane < 16) {
    float m = mask[j0 + lane];
    scores[(size_t)(g * 2 + 0) * LCTX + j0 + lane] = acc[0] * sc + m;
    scores[(size_t)(g * 2 + 1) * LCTX + j0 + lane] = acc[1] * sc + m;
  }
}

// ---------------- softmax statistics per q-head (also zeroes attn accumulators) ----------------
__global__ void softmax_reduce_kernel(const float* __restrict__ scores,
                                      float* __restrict__ smax, float* __restrict__ sinv,
                                      float* __restrict__ attn) {
  __shared__ float red[8];
  __shared__ float bmax;
  int h = blockIdx.x, t = threadIdx.x;
  const float* s = scores + (size_t)h * LCTX;
  float m = -1e30f;
  for (int j = t; j < LCTX; j += 256) m = fmaxf(m, s[j]);
  m = wave_reduce_max(m);
  if ((t & 31) == 0) red[t >> 5] = m;
  __syncthreads();
  if (t == 0) {
    float mm = red[0];
    #pragma unroll
    for (int k = 1; k < 8; ++k) mm = fmaxf(mm, red[k]);
    bmax = mm;
  }
  __syncthreads();
  float M = bmax;
  float ssum = 0.f;
  for (int j = t; j < LCTX; j += 256) ssum += expf(s[j] - M);
  ssum = wave_reduce_sum(ssum);
  __syncthreads();
  if ((t & 31) == 0) red[t >> 5] = ssum;
  __syncthreads();
  if (t == 0) {
    float ts = 0.f;
    #pragma unroll
    for (int k = 0; k < 8; ++k) ts += red[k];
    smax[h] = M;
    sinv[h] = 1.f / ts;
  }
  if (t < 128) attn[h * 128 + t] = 0.f;
}

// ---------------- fused p@V + V-cache copy-out ----------------
// grid = KV*64 blocks, 128 threads (thread = dim d); chunk of 128 positions.
__global__ void attnv_copyv_kernel(const float* __restrict__ pastV, const float* __restrict__ newv,
                                   const float* __restrict__ scores,
                                   const float* __restrict__ smax, const float* __restrict__ sinv,
                                   const int* __restrict__ cache_pos,
                                   float* __restrict__ outV, float* __restrict__ attn) {
  int b = blockIdx.x;
  int g = b >> 6, c = b & 63;
  int d = threadIdx.x;
  int pos = cache_pos[0];
  int j0 = c * 128;
  const float* Vg = pastV + (size_t)g * LCTX * D;
  float*       Og = outV  + (size_t)g * LCTX * D;
  int h0 = g * 2, h1 = g * 2 + 1;
  float m0 = smax[h0], i0 = sinv[h0];
  float m1 = smax[h1], i1 = sinv[h1];
  const float* s0 = scores + (size_t)h0 * LCTX;
  const float* s1 = scores + (size_t)h1 * LCTX;
  float a0 = 0.f, a1 = 0.f;
  for (int j = j0; j < j0 + 128; ++j) {
    float v = (j == pos) ? newv[g * D + d] : Vg[(size_t)j * D + d];
    Og[(size_t)j * D + d] = v;                       // fused cache copy-out
    a0 += expf(s0[j] - m0) * i0 * v;
    a1 += expf(s1[j] - m1) * i1 * v;
  }
  atomicAdd(&attn[h0 * 128 + d], a0);
  atomicAdd(&attn[h1 * 128 + d], a1);
}

// ---------------- host launch ----------------
extern "C" void kernel_launch(void* const* d_in, const int* in_sizes, int n_in,
                              void* d_out, int out_size, void* d_ws, size_t ws_size,
                              hipStream_t stream) {
  (void)in_sizes; (void)n_in; (void)out_size; (void)ws_size;
  const float* emb  = (const float*)d_in[0];
  const int*   cpos = (const int*)d_in[1];
  const float* mask = (const float*)d_in[2];
  const float* pK   = (const float*)d_in[3];
  const float* pV   = (const float*)d_in[4];
  const float* ln1  = (const float*)d_in[5];
  const float* qw   = (const float*)d_in[6];
  const float* kw   = (const float*)d_in[7];
  const float* vw   = (const float*)d_in[8];
  const float* qn   = (const float*)d_in[9];
  const float* kn   = (const float*)d_in[10];
  const float* ow   = (const float*)d_in[11];
  const float* ln2  = (const float*)d_in[12];
  const float* gw   = (const float*)d_in[13];
  const float* uw   = (const float*)d_in[14];
  const float* dw   = (const float*)d_in[15];
  const float* nw   = (const float*)d_in[16];

  float* out_h  = (float*)d_out;
  float* out_pk = out_h + HID;
  float* out_pv = out_pk + (size_t)NL * KV * LCTX * D;

  float* W = (float*)d_ws;
  float* ws_h      = W;                    // 1024
  float* ws_n      = W + 1024;             // 1024
  float* ws_n2     = W + 2048;             // 1024
  float* ws_qraw   = W + 3072;             // 2048
  float* ws_kraw   = W + 5120;             // 1024
  float* ws_q      = W + 6144;             // 2048
  float* ws_newk   = W + 8192;             // 1024
  float* ws_newv   = W + 9216;             // 1024
  float* ws_attn   = W + 10240;            // 2048
  float* ws_act    = W + 12288;            // 3072
  float* ws_smax   = W + 15360;            // 16
  float* ws_sinv   = W + 15376;            // 16
  float* ws_scores = W + 15392;            // H*L = 131072

  copy_kernel<<<4, 256, 0, stream>>>(emb, ws_h, HID);

  for (int i = 0; i < NL; ++i) {
    const size_t cachL = (size_t)KV * LCTX * D;
    rmsnorm_kernel<<<1, 256, 0, stream>>>(ws_h, ln1 + (size_t)i * HID, ws_n, HID);

    gemv_kernel<<<256, 256, 0, stream>>>(qw + (size_t)i * (H * D) * HID, ws_n, ws_qraw, H * D, HID);
    gemv_kernel<<<128, 256, 0, stream>>>(kw + (size_t)i * (KV * D) * HID, ws_n, ws_kraw, KV * D, HID);
    gemv_kernel<<<128, 256, 0, stream>>>(vw + (size_t)i * (KV * D) * HID, ws_n, ws_newv, KV * D, HID);

    qknorm_rope_kernel<<<24, 128, 0, stream>>>(ws_qraw, ws_kraw,
                                               qn + (size_t)i * D, kn + (size_t)i * D,
                                               cpos, ws_q, ws_newk);

    score_copyk_kernel<<<1024, 128, 0, stream>>>(pK + (size_t)i * cachL, ws_newk, ws_q, mask,
                                                 cpos, out_pk + (size_t)i * cachL, ws_scores);

    softmax_reduce_kernel<<<16, 256, 0, stream>>>(ws_scores, ws_smax, ws_sinv, ws_attn);

    attnv_copyv_kernel<<<512, 128, 0, stream>>>(pV + (size_t)i * cachL, ws_newv, ws_scores,
                                                ws_smax, ws_sinv, cpos,
                                                out_pv + (size_t)i * cachL, ws_attn);

    gemv_residual_kernel<<<128, 256, 0, stream>>>(ow + (size_t)i * HID * (H * D), ws_attn, ws_h,
                                                  HID, H * D);

    rmsnorm_kernel<<<1, 256, 0, stream>>>(ws_h, ln2 + (size_t)i * HID, ws_n2, HID);

    gemv_gateup_kernel<<<384, 256, 0, stream>>>(gw + (size_t)i * FF * HID,
                                                uw + (size_t)i * FF * HID,
                                                ws_n2, ws_act, FF, HID);

    gemv_residual_kernel<<<128, 256, 0, stream>>>(dw + (size_t)i * HID * FF, ws_act, ws_h,
                                                  HID, FF);
  }

  rmsnorm_kernel<<<1, 256, 0, stream>>>(ws_h, nw, out_h, HID);
}